// dwt_concat_38371237822840
// MI455X (gfx1250) — compile-verified
//
#include <hip/hip_runtime.h>
#include <cstdint>

// Shapes: x1,x2 = (8,64,256,256) f32 ; out = (8,128,256,256) f32
// out[b, 0:64]   = haar_fuse_reconstruct(x1, x2)
// out[b, 64:128] = x2[b]
//
// Memory-bound: 512 MiB total traffic -> ~22 us floor at 23.3 TB/s.

typedef float v4f __attribute__((ext_vector_type(4)));

struct Tile2x2 { float tl, tr, bl, br; };

__device__ __forceinline__ float ffm_sel(float p, float q) {
    return (__builtin_fabsf(p) >= __builtin_fabsf(q)) ? p : q;
}

// One fused 2x2 Haar block: inputs are the 2x2 tiles of x1 and x2
// (tl,tr,bl,br), returns the reconstructed 2x2 tile.
__device__ __forceinline__ Tile2x2 haar_block(
    float a1, float b1, float c1, float d1,   // x1: tl,tr,bl,br
    float a2, float b2, float c2, float d2)   // x2: tl,tr,bl,br
{
    // Forward DWT (the /2 on inputs folded into the *0.5f)
    float A1 = ( a1 + b1 + c1 + d1) * 0.5f;
    float B1 = (-a1 + b1 - c1 + d1) * 0.5f;
    float C1 = (-a1 - b1 + c1 + d1) * 0.5f;
    float D1 = ( a1 - b1 - c1 + d1) * 0.5f;
    float A2 = ( a2 + b2 + c2 + d2) * 0.5f;
    float B2 = (-a2 + b2 - c2 + d2) * 0.5f;
    float C2 = (-a2 - b2 + c2 + d2) * 0.5f;
    float D2 = ( a2 - b2 - c2 + d2) * 0.5f;
    // Fusion
    float fa = (A1 + A2) * 0.5f;
    float fb = ffm_sel(B1, B2);
    float fc = ffm_sel(C1, C2);
    float fd = ffm_sel(D1, D2);
    // Inverse DWT (the final x2 scaling folds /4 -> /2)
    Tile2x2 o;
    o.tl = (fa - fb - fc + fd) * 0.5f;
    o.tr = (fa + fb - fc - fd) * 0.5f;
    o.bl = (fa - fb + fc - fd) * 0.5f;
    o.br = (fa + fb + fc + fd) * 0.5f;
    return o;
}

// Grid layout:
//   blocks [0, 16384)      : reconstruction half. 4,194,304 threads, each
//                            handles 2 rows x 4 cols (two 2x2 tiles).
//   blocks [16384, 49152)  : copy x2 -> out upper channels via CDNA5
//                            async LDS streaming path. 8,388,608 threads,
//                            one b128 each.
__global__ __launch_bounds__(256) void dwt_fuse_concat_kernel(
    const float* __restrict__ x1, const float* __restrict__ x2,
    float* __restrict__ out)
{
    __shared__ v4f sbuf[256];   // 4 KB staging for the async-copy half
    const unsigned tid = threadIdx.x;
    const unsigned bid = blockIdx.x;

    if (bid < 16384u) {
        // ---------------- reconstruction half ----------------
        unsigned t     = bid * 256u + tid;     // [0, 4194304)
        unsigned j     = t & 63u;              // quad-column (4 floats)
        unsigned i     = (t >> 6) & 127u;      // row-pair
        unsigned plane = t >> 13;              // b*64 + c, [0, 512)
        unsigned b     = plane >> 6;

        size_t in_off  = (size_t)plane * 65536u + (size_t)(2u * i) * 256u + 4u * j;
        size_t out_off = (size_t)(plane + b * 64u) * 65536u
                       + (size_t)(2u * i) * 256u + 4u * j;

        // x1 is dead after this kernel -> non-temporal loads.
        v4f r0a = __builtin_nontemporal_load((const v4f*)(x1 + in_off));
        v4f r1a = __builtin_nontemporal_load((const v4f*)(x1 + in_off + 256));
        // x2 is re-read by the concat half -> regular (RT) loads, let L2 keep it.
        v4f r0b = *(const v4f*)(x2 + in_off);
        v4f r1b = *(const v4f*)(x2 + in_off + 256);

        // tile 0: cols (4j, 4j+1) ; tile 1: cols (4j+2, 4j+3)
        Tile2x2 t0 = haar_block(r0a.x, r0a.y, r1a.x, r1a.y,
                                r0b.x, r0b.y, r1b.x, r1b.y);
        Tile2x2 t1 = haar_block(r0a.z, r0a.w, r1a.z, r1a.w,
                                r0b.z, r0b.w, r1b.z, r1b.w);

        v4f o0, o1;
        o0.x = t0.tl; o0.y = t0.tr; o0.z = t1.tl; o0.w = t1.tr;
        o1.x = t0.bl; o1.y = t0.br; o1.z = t1.bl; o1.w = t1.br;

        // Output is streamed out, never re-read -> non-temporal stores.
        __builtin_nontemporal_store(o0, (v4f*)(out + out_off));
        __builtin_nontemporal_store(o1, (v4f*)(out + out_off + 256));
    } else {
        // ---------------- concat-copy half: out[b, 64:128] = x2[b] ----------
        // Pure streaming copy: use the CDNA5 async LDS path
        // (global -> LDS -> global, no VGPR round-trip, tracked by ASYNCcnt).
        unsigned q = (bid - 16384u) * 256u + tid;   // [0, 8388608) float4 units
        unsigned b = q >> 20;                       // batch (1,048,576 f4/chunk)
        unsigned r = q & 1048575u;

        const v4f* src = (const v4f*)x2 + ((size_t)b * 1048576u + r);
        v4f*       dst = (v4f*)out      + ((size_t)b * 2097152u + 1048576u + r);

        unsigned lds_addr = (unsigned)(uintptr_t)(&sbuf[tid]); // low 32b = LDS offset
        unsigned long long gsrc = (unsigned long long)(uintptr_t)src;
        unsigned long long gdst = (unsigned long long)(uintptr_t)dst;

        asm volatile(
            "global_load_async_to_lds_b128 %0, %1, off\n\t"
            "s_wait_asynccnt 0\n\t"
            "global_store_async_from_lds_b128 %2, %0, off\n\t"
            "s_wait_asynccnt 0"
            :
            : "v"(lds_addr), "v"(gsrc), "v"(gdst)
            : "memory");
    }
}

extern "C" void kernel_launch(void* const* d_in, const int* in_sizes, int n_in,
                              void* d_out, int out_size, void* d_ws, size_t ws_size,
                              hipStream_t stream) {
    const float* x1 = (const float*)d_in[0];
    const float* x2 = (const float*)d_in[1];
    float* out = (float*)d_out;

    // 16384 reconstruction blocks + 32768 async-copy blocks
    dim3 grid(49152), block(256);
    hipLaunchKernelGGL(dwt_fuse_concat_kernel, grid, block, 0, stream,
                       x1, x2, out);
}